// CRFLayer_88622355186398
// MI455X (gfx1250) — compile-verified
//
#include <hip/hip_runtime.h>
#include <stdint.h>
#include <stddef.h>

// ---------------------------------------------------------------------------
// CRF layer for gfx1250 (MI455X).
//
// Forward all-path scan rewritten in the exp-domain so the per-step
// log-sum-exp "matmul" becomes a real f32 GEMM on V_WMMA_F32_16X16X4_F32:
//     new_alpha[i][b] = feat[i][b] + m_b + log( sum_j E[i][j]*exp(alpha[j][b]-m_b) )
// with E = exp(transition) precomputed once into LDS.
//
// alpha is kept transposed in the WMMA C/D register layout:
//     acc[mt][r] <-> (tag = 16*mt + 8*(lane>>4) + r, batch = lane&15)
// Each wave owns 16 batch columns; 4 waves / block; 16 blocks cover B=1024.
// Feats are staged to LDS double-buffered via async-to-LDS (ASYNCcnt).
// ---------------------------------------------------------------------------

typedef __attribute__((ext_vector_type(2))) float v2f;
typedef __attribute__((ext_vector_type(8))) float v8f;
typedef __attribute__((ext_vector_type(4))) int   v4i;

typedef __attribute__((address_space(1))) v4i* gv4i_p;  // global-AS vector ptr
typedef __attribute__((address_space(3))) v4i* lv4i_p;  // LDS-AS vector ptr

#define SEQ_L     512
#define BATCH_N   1024
#define NTAG      64
#define START_TAG 62
#define END_TAG   63
#define NEGV      (-10000.0f)

#define ESTRIDE 68   // padded row stride (floats) for E / vexp tiles: bank-conflict free (4*col + t)
#define FSTRIDE 72   // padded row stride (floats) for feat staging: rows stay 16B aligned for b128

#ifndef HAVE_ASYNC
#if defined(__has_builtin)
#  if __has_builtin(__builtin_amdgcn_global_load_async_to_lds_b128) && \
      __has_builtin(__builtin_amdgcn_s_wait_asynccnt)
#    define HAVE_ASYNC 1
#  endif
#endif
#endif
#ifndef HAVE_ASYNC
#define HAVE_ASYNC 0
#endif

#ifndef HAVE_WMMA_F32X4
#if defined(__has_builtin)
#  if __has_builtin(__builtin_amdgcn_wmma_f32_16x16x4_f32)
#    define HAVE_WMMA_F32X4 1
#  endif
#endif
#endif
#ifndef HAVE_WMMA_F32X4
#define HAVE_WMMA_F32X4 0
#endif

// Stage one 16x64 f32 feat tile (global, row stride 64) into LDS (row stride
// FSTRIDE). 256 16-byte chunks spread across 32 lanes -> 8 b128 transfers.
__device__ __forceinline__
void stage_tile(const float* __restrict__ g, float* __restrict__ l, int lane)
{
#pragma unroll
  for (int k = 0; k < 8; ++k) {
    const int c   = lane + 32 * k;      // chunk id 0..255
    const int row = c >> 4;             // batch row 0..15
    const int c4  = (c & 15) << 2;      // float col, multiple of 4
    const float* gp = g + row * NTAG + c4;
    float*       lp = l + row * FSTRIDE + c4;
#if HAVE_ASYNC
    float* gnc = const_cast<float*>(gp);
    __builtin_amdgcn_global_load_async_to_lds_b128(
        (gv4i_p)gnc, (lv4i_p)lp, 0, 0);
#else
    const float4 v = *(const float4*)gp;
    *(float4*)lp = v;
#endif
  }
}

__launch_bounds__(128)
__global__ void crf_allpath_kernel(const float* __restrict__ feats,
                                   const float* __restrict__ mask,
                                   const float* __restrict__ transition,
                                   float* __restrict__ allpath)
{
  __shared__ float sE[NTAG * ESTRIDE];        // E = exp(transition), padded rows
  __shared__ float sV[4][16 * ESTRIDE];       // per-wave exp(alpha - m) transpose buffer
  __shared__ float sF[2][4][16 * FSTRIDE];    // per-wave double-buffered feat tiles

  const int tid  = threadIdx.x;
  const int wave = tid >> 5;
  const int lane = tid & 31;
  const int col  = lane & 15;                 // batch column within wave tile
  const int half = lane >> 4;                 // lane-half selects tag +8 / K +2
  const int b0   = blockIdx.x * 64 + wave * 16;

  // ---- Build E = exp(transition) in LDS (rows padded to ESTRIDE) ----
  for (int i = tid; i < NTAG * NTAG; i += 128) {
    const int r = i >> 6, c = i & 63;
    sE[r * ESTRIDE + c] = __expf(transition[i]);
  }
  __syncthreads();

  float* sVw = &sV[wave][0];

  // ---- alpha0: NEG everywhere, 0 at START (transposed WMMA C layout) ----
  v8f acc[4];
#pragma unroll
  for (int mt = 0; mt < 4; ++mt)
#pragma unroll
    for (int r = 0; r < 8; ++r) {
      const int tag = 16 * mt + 8 * half + r;
      acc[mt][r] = (tag == START_TAG) ? 0.0f : NEGV;
    }

  int cur = 0;
  stage_tile(feats + (size_t)b0 * NTAG, &sF[0][wave][0], lane);

  for (int l = 0; l < SEQ_L; ++l) {
    const int nxt = cur ^ 1;
    // Always issue exactly 8 async ops (wrap on last step) so the
    // "s_wait_asynccnt 8" below always means "current buffer landed".
    const int lpre = (l + 1 < SEQ_L) ? (l + 1) : 0;
    stage_tile(feats + ((size_t)lpre * BATCH_N + b0) * NTAG, &sF[nxt][wave][0], lane);

    // 1) m_b = max over all 64 tags of alpha[:,b]  (local 32 + xor-16 shuffle)
    float mb = -3.0e38f;
#pragma unroll
    for (int mt = 0; mt < 4; ++mt)
#pragma unroll
      for (int r = 0; r < 8; ++r) mb = fmaxf(mb, acc[mt][r]);
    mb = fmaxf(mb, __shfl_xor(mb, 16));

    // 2) vexp = exp(alpha - m_b) -> LDS, stored [batch col][tag] (transpose)
#pragma unroll
    for (int mt = 0; mt < 4; ++mt)
#pragma unroll
      for (int r = 0; r < 8; ++r)
        sVw[col * ESTRIDE + 16 * mt + 8 * half + r] = __expf(acc[mt][r] - mb);
    __builtin_amdgcn_wave_barrier();   // keep DS stores before the DS loads below

    // 3) B fragments: b64 per K-step; lane holds vexp[4s+2*half+{0,1}][col]
    v2f bf[16];
#pragma unroll
    for (int s = 0; s < 16; ++s)
      bf[s] = *(const v2f*)&sVw[col * ESTRIDE + 4 * s + 2 * half];

    // 4) D = E x Vexp : 4 M-tiles x 16 K-steps of V_WMMA_F32_16X16X4_F32
    v8f d[4];
#pragma unroll
    for (int mt = 0; mt < 4; ++mt) {
      v8f t = {0.f, 0.f, 0.f, 0.f, 0.f, 0.f, 0.f, 0.f};
#pragma unroll
      for (int s = 0; s < 16; ++s) {
        const v2f a = *(const v2f*)&sE[(16 * mt + col) * ESTRIDE + 4 * s + 2 * half];
#if HAVE_WMMA_F32X4
        t = __builtin_amdgcn_wmma_f32_16x16x4_f32(
                false, a, false, bf[s], (short)0, t, false, false);
#else
        // placeholder so the TU still compiles if the builtin is absent
#pragma unroll
        for (int r = 0; r < 8; ++r) t[r] += a[0] * bf[s][0] + a[1] * bf[s][1];
#endif
      }
      d[mt] = t;
    }

    // 5) current feat tile is ready once at most the 8 newest async ops remain
#if HAVE_ASYNC
    __builtin_amdgcn_s_wait_asynccnt(8);
#endif
    __builtin_amdgcn_wave_barrier();

    // 6) new_alpha = feat + m_b + log(dot), masked blend with old alpha
    const float mk = mask[(size_t)l * BATCH_N + b0 + col];
    const float* sFw = &sF[cur][wave][0];
#pragma unroll
    for (int mt = 0; mt < 4; ++mt)
#pragma unroll
      for (int r = 0; r < 8; ++r) {
        const int tag  = 16 * mt + 8 * half + r;
        const float f  = sFw[col * FSTRIDE + tag];
        const float nv = f + mb + __logf(d[mt][r]);
        acc[mt][r] = mk * nv + (1.0f - mk) * acc[mt][r];
      }
    __builtin_amdgcn_wave_barrier();
    cur = nxt;
  }

  // ---- finalize: alpha += transition[END][tag]; logsumexp over tags ----
#pragma unroll
  for (int mt = 0; mt < 4; ++mt)
#pragma unroll
    for (int r = 0; r < 8; ++r)
      acc[mt][r] += transition[END_TAG * NTAG + 16 * mt + 8 * half + r];

  float M = -3.0e38f;
#pragma unroll
  for (int mt = 0; mt < 4; ++mt)
#pragma unroll
    for (int r = 0; r < 8; ++r) M = fmaxf(M, acc[mt][r]);
  M = fmaxf(M, __shfl_xor(M, 16));

  float S = 0.0f;
#pragma unroll
  for (int mt = 0; mt < 4; ++mt)
#pragma unroll
    for (int r = 0; r < 8; ++r) S += __expf(acc[mt][r] - M);
  S += __shfl_xor(S, 16);

  if (lane < 16) allpath[b0 + col] = M + __logf(S);
}

// ---------------------------------------------------------------------------
// Gold-path score (gather) + final subtraction. One thread per batch element.
// ---------------------------------------------------------------------------
__launch_bounds__(256)
__global__ void crf_realpath_kernel(const float* __restrict__ feats,
                                    const int*   __restrict__ tags,
                                    const float* __restrict__ mask,
                                    const float* __restrict__ transition,
                                    const float* __restrict__ allpath,
                                    float* __restrict__ out)
{
  const int b = blockIdx.x * blockDim.x + threadIdx.x;
  if (b >= BATCH_N) return;

  int   prev  = START_TAG;
  float score = 0.0f;
  int   len   = 0;
  for (int l = 0; l < SEQ_L; ++l) {
    const int   t = tags[(size_t)l * BATCH_N + b];
    const float m = mask[(size_t)l * BATCH_N + b];
    const float emit = feats[((size_t)l * BATCH_N + b) * NTAG + t];
    score += (emit + transition[t * NTAG + prev]) * m;
    len   += (int)m;
    prev   = t;
  }
  const int last = (len == 0) ? START_TAG : tags[(size_t)(len - 1) * BATCH_N + b];
  score += transition[END_TAG * NTAG + last];

  out[b] = allpath[b] - score;
}

extern "C" void kernel_launch(void* const* d_in, const int* in_sizes, int n_in,
                              void* d_out, int out_size, void* d_ws, size_t ws_size,
                              hipStream_t stream) {
  (void)in_sizes; (void)n_in; (void)out_size; (void)ws_size;
  const float* feats      = (const float*)d_in[0];   // (512,1024,64) f32
  const int*   tags       = (const int*)  d_in[1];   // (512,1024) i32
  const float* mask       = (const float*)d_in[2];   // (512,1024) f32
  const float* transition = (const float*)d_in[3];   // (64,64) f32
  float*       out        = (float*)d_out;           // (1024,) f32
  float*       allpath    = (float*)d_ws;            // scratch: 1024 f32

  crf_allpath_kernel<<<BATCH_N / 64, 128, 0, stream>>>(feats, mask, transition, allpath);
  crf_realpath_kernel<<<BATCH_N / 256, 256, 0, stream>>>(feats, tags, mask, transition,
                                                         allpath, out);
}